// GNN_25134148616988
// MI455X (gfx1250) — compile-verified
//
#include <hip/hip_runtime.h>
#include <hip/hip_bf16.h>

// ---------------------------------------------------------------------------
// GATv2 GNN forward for MI455X (gfx1250, wave32).
// Dense projections: V_WMMA_F32_16X16X4_F32 (exact fp32 matrix pipe), with
// the A tile staged in LDS (shared by all 4 waves of the block) and NT
// 16x16 C tiles per wave so one A fragment feeds NT back-to-back WMMAs.
// Irregular edge ops: wave-per-edge gathers, wave32 shuffle reductions,
// ordered-uint atomic max for segment softmax, f32 atomic scatter-add.
// ---------------------------------------------------------------------------

#define N_NODES 16384
#define N_EDGES 262144
#define N_GRAPHS 512
#define FEAT 64
#define EDGE_DIM 16
#define HEADS 4
#define EMB 128
#define HC 512
#define DENSE 1024
#define BN_EPS 1e-5f
#define NEG_SLOPE 0.2f

typedef __attribute__((ext_vector_type(2))) float v2f;
typedef __attribute__((ext_vector_type(8))) float v8f;

// ---- monotone float <-> uint mapping for atomic max on f32 ---------------
__device__ __forceinline__ unsigned fkey(float f) {
    unsigned u = __float_as_uint(f);
    return (u & 0x80000000u) ? ~u : (u | 0x80000000u);
}
__device__ __forceinline__ float fdecode(unsigned u) {
    return (u & 0x80000000u) ? __uint_as_float(u & 0x7fffffffu)
                             : __uint_as_float(~u);
}

// ---------------------------------------------------------------------------
// WMMA fp32 GEMM: C[M,N] = A[M,K] @ B[K,N]  (+ epilogue)
// Block = 128 threads (4 waves). Each block owns one 16-row M tile, staged
// once into LDS; each wave computes NT adjacent 16x16 C tiles so each A
// fragment (ds_load_b64) feeds NT consecutive v_wmma ops.
// Grid must tile M,N exactly (EXEC all-1s required for WMMA); K % 4 == 0.
// mode 0: none; 1: +bias[n]; 2: bn(relu(c + bias[n]))
// Dynamic LDS: 16*(K+4)*4 bytes (stride padded for bank spread + 16B align).
// ---------------------------------------------------------------------------
template <int NT>
__global__ __launch_bounds__(128)
void wmma_gemm_f32(const float* __restrict__ A, const float* __restrict__ B,
                   float* __restrict__ C, int M, int N, int K,
                   const float* __restrict__ bias,
                   const float* __restrict__ bn_g, const float* __restrict__ bn_b,
                   const float* __restrict__ bn_mean, const float* __restrict__ bn_var,
                   int mode)
{
    extern __shared__ float lds_a[];
    const int tid  = threadIdx.x;
    const int lane = tid & 31;
    const int wave = tid >> 5;
    const int tm   = blockIdx.x * 16;
    const int tn   = (blockIdx.y * 4 + wave) * (NT * 16);
    const int half = lane >> 4;     // 0: K pair {0,1}; 1: K pair {2,3}
    const int l15  = lane & 15;
    const int lstride = K + 4;      // dword stride: bank-spread + float4 align

    // ---- cooperative stage of the 16 x K A tile into LDS (float4 chunks)
    const int k4n = K >> 2;
    for (int r = 0; r < 16; ++r) {
        const float4* srcv = (const float4*)(A + (long)(tm + r) * K);
        float4*       dstv = (float4*)(lds_a + r * lstride);
        for (int k4 = tid; k4 < k4n; k4 += 128) dstv[k4] = srcv[k4];
    }
    __syncthreads();

    v8f acc[NT];
#pragma unroll
    for (int t = 0; t < NT; ++t) acc[t] = (v8f){};

    const float* ap = lds_a + l15 * lstride + half * 2;          // LDS reads
    const float* bp = B + (long)(half * 2) * N + tn + l15;       // global reads
    const long bstep = 4L * N;

    for (int k0 = 0; k0 < K; k0 += 4) {
        v2f a;
        a.x = ap[0];
        a.y = ap[1];
        __builtin_prefetch(bp + 2 * bstep, 0, 1);
#pragma unroll
        for (int t = 0; t < NT; ++t) {
            v2f b;
            b.x = bp[t * 16];
            b.y = bp[t * 16 + N];
            acc[t] = __builtin_amdgcn_wmma_f32_16x16x4_f32(
                /*neg_a=*/false, a, /*neg_b=*/false, b,
                /*c_mod=*/(short)0, acc[t], /*reuse_a=*/false, /*reuse_b=*/false);
        }
        ap += 4;
        bp += bstep;
    }

    // ---- epilogue + store (C 16x16 layout: vgpr r -> row, lane -> col)
    const int row0 = tm + half * 8;
#pragma unroll
    for (int t = 0; t < NT; ++t) {
        const int col = tn + t * 16 + l15;
        float sc = 0.f, sb = 0.f, bi = 0.f;
        if (mode == 1) {
            bi = bias[col];
        } else if (mode == 2) {
            bi = bias[col];
            sc = rsqrtf(bn_var[col] + BN_EPS) * bn_g[col];
            sb = bn_b[col] - bn_mean[col] * sc;
        }
#pragma unroll
        for (int r = 0; r < 8; ++r) {
            float v = acc[t][r];
            if (mode == 1)      v += bi;
            else if (mode == 2) v = fmaxf(v + bi, 0.0f) * sc + sb;
            C[(long)(row0 + r) * N + col] = v;
        }
    }
}

// ---------------------------------------------------------------------------
// Per-layer init: out accumulator = conv bias (broadcast), reset softmax state
// ---------------------------------------------------------------------------
__global__ __launch_bounds__(256)
void init_layer_kernel(float* __restrict__ acc, const float* __restrict__ cbias,
                       unsigned* __restrict__ nmax, float* __restrict__ nden)
{
    const int t = blockIdx.x * 256 + threadIdx.x;       // N_NODES*HC threads
    acc[t] = cbias[t & (HC - 1)];
    if (t < N_NODES * HEADS) { nmax[t] = 0u; nden[t] = 0.0f; }
}

// ---------------------------------------------------------------------------
// Edge pass 1: wave per edge. Computes lin_edge(e) on the fly, builds
// m = leaky_relu(xl[src]+xr[dst]+e), logits[h] = <m_h, att_h>, atomic segmax.
// ---------------------------------------------------------------------------
__global__ __launch_bounds__(256)
void edge_logits_kernel(const int* __restrict__ ei, const float* __restrict__ eattr,
                        const float* __restrict__ XL, const float* __restrict__ XR,
                        const float* __restrict__ We, const float* __restrict__ att,
                        float* __restrict__ logits, unsigned* __restrict__ nmax)
{
    const int e    = blockIdx.x * 8 + (threadIdx.x >> 5);
    const int lane = threadIdx.x & 31;
    const int src  = ei[e];
    const int dst  = ei[N_EDGES + e];

    float ea[EDGE_DIM];
#pragma unroll
    for (int k = 0; k < EDGE_DIM; ++k) ea[k] = eattr[(long)e * EDGE_DIM + k];

    float hsum[HEADS] = {0.f, 0.f, 0.f, 0.f};
#pragma unroll
    for (int j = 0; j < 16; ++j) {
        const int col = lane + 32 * j;                  // 0..511, head = col>>7
        float ep = 0.0f;
#pragma unroll
        for (int k = 0; k < EDGE_DIM; ++k) ep += ea[k] * We[k * HC + col];
        float m = XL[(long)src * HC + col] + XR[(long)dst * HC + col] + ep;
        m = (m < 0.0f) ? NEG_SLOPE * m : m;
        hsum[j >> 2] += m * att[col];                   // att flat [H*EMB] == [col]
    }
#pragma unroll
    for (int h = 0; h < HEADS; ++h) {
        float s = hsum[h];
        for (int off = 16; off > 0; off >>= 1) s += __shfl_down(s, off, 32);
        if (lane == 0) {
            logits[(long)e * HEADS + h] = s;
            atomicMax(&nmax[dst * HEADS + h], fkey(s));
        }
    }
}

// ---------------------------------------------------------------------------
// Edge pass 2: ex = exp(logit - max[dst]); accumulate denominator
// ---------------------------------------------------------------------------
__global__ __launch_bounds__(256)
void edge_exp_kernel(const int* __restrict__ ei, const float* __restrict__ logits,
                     const unsigned* __restrict__ nmax, float* __restrict__ alpha,
                     float* __restrict__ nden)
{
    const int t = blockIdx.x * 256 + threadIdx.x;       // N_EDGES*HEADS
    const int e = t >> 2, h = t & 3;
    const int dst = ei[N_EDGES + e];
    const float ex = __expf(logits[t] - fdecode(nmax[dst * HEADS + h]));
    alpha[t] = ex;
    atomicAdd(&nden[dst * HEADS + h], ex);
}

// ---------------------------------------------------------------------------
// Edge pass 3: wave per edge, scatter alpha * xl[src] into dst accumulator
// ---------------------------------------------------------------------------
__global__ __launch_bounds__(256)
void edge_scatter_kernel(const int* __restrict__ ei, const float* __restrict__ XL,
                         const float* __restrict__ alpha, const float* __restrict__ nden,
                         float* __restrict__ acc)
{
    const int e    = blockIdx.x * 8 + (threadIdx.x >> 5);
    const int lane = threadIdx.x & 31;
    const int src  = ei[e];
    const int dst  = ei[N_EDGES + e];
    float w[HEADS];
#pragma unroll
    for (int h = 0; h < HEADS; ++h)
        w[h] = alpha[(long)e * HEADS + h] / nden[dst * HEADS + h];
#pragma unroll
    for (int j = 0; j < 16; ++j) {
        const int col = lane + 32 * j;
        atomicAdd(&acc[(long)dst * HC + col], w[j >> 2] * XL[(long)src * HC + col]);
    }
}

// ---------------------------------------------------------------------------
// Pooling
// ---------------------------------------------------------------------------
__global__ __launch_bounds__(256)
void pool_init_kernel(float* __restrict__ hbuf, const float* __restrict__ molc,
                      const float* __restrict__ cenc, unsigned* __restrict__ gmax,
                      float* __restrict__ gden)
{
    const int t = blockIdx.x * 256 + threadIdx.x;
    if (t < N_GRAPHS * (EMB + 2)) {
        const int g = t / (EMB + 2), c = t % (EMB + 2);
        hbuf[t] = (c == EMB) ? molc[g] : (c == EMB + 1) ? cenc[g] : 0.0f;
    }
    if (t < N_GRAPHS) { gmax[t] = 0u; gden[t] = 0.0f; }
}

__global__ __launch_bounds__(256)
void gate_kernel(const float* __restrict__ X, const int* __restrict__ node_index,
                 const int* __restrict__ batch_index, const float* __restrict__ gw,
                 const float* __restrict__ gb, float* __restrict__ gval,
                 unsigned* __restrict__ gmax)
{
    const int i    = blockIdx.x * 8 + (threadIdx.x >> 5);   // 8192 selected nodes
    const int lane = threadIdx.x & 31;
    const int node = node_index[i];
    float s = 0.0f;
    for (int c = lane; c < EMB; c += 32) s += X[(long)node * EMB + c] * gw[c];
    for (int off = 16; off > 0; off >>= 1) s += __shfl_down(s, off, 32);
    if (lane == 0) {
        s += gb[0];
        gval[i] = s;
        atomicMax(&gmax[batch_index[node]], fkey(s));
    }
}

__global__ __launch_bounds__(256)
void gate_exp_kernel(const int* __restrict__ node_index, const int* __restrict__ batch_index,
                     const float* __restrict__ gval, const unsigned* __restrict__ gmax,
                     float* __restrict__ gex, float* __restrict__ gden)
{
    const int i = blockIdx.x * 256 + threadIdx.x;           // 8192
    const int b = batch_index[node_index[i]];
    const float ex = __expf(gval[i] - fdecode(gmax[b]));
    gex[i] = ex;
    atomicAdd(&gden[b], ex);
}

__global__ __launch_bounds__(256)
void pool_scatter_kernel(const float* __restrict__ X, const int* __restrict__ node_index,
                         const int* __restrict__ batch_index, const float* __restrict__ gex,
                         const float* __restrict__ gden, float* __restrict__ hbuf)
{
    const int i    = blockIdx.x * 8 + (threadIdx.x >> 5);
    const int lane = threadIdx.x & 31;
    const int node = node_index[i];
    const int b    = batch_index[node];
    const float w  = gex[i] / gden[b];
    for (int c = lane; c < EMB; c += 32)
        atomicAdd(&hbuf[(long)b * (EMB + 2) + c], w * X[(long)node * EMB + c]);
}

// ---------------------------------------------------------------------------
// Small dense layers (MLP head): one thread per output element
// ---------------------------------------------------------------------------
__global__ __launch_bounds__(256)
void fc_kernel(const float* __restrict__ in, const float* __restrict__ W,
               const float* __restrict__ bias, float* __restrict__ out,
               int M, int N, int K, int do_relu)
{
    const int t = blockIdx.x * 256 + threadIdx.x;
    if (t >= M * N) return;
    const int m = t / N, n = t % N;
    float s = bias[n];
    for (int k = 0; k < K; ++k) s += in[(long)m * K + k] * W[(long)k * N + n];
    out[t] = do_relu ? fmaxf(s, 0.0f) : s;
}

// ---------------------------------------------------------------------------
// Host launcher
// ---------------------------------------------------------------------------
extern "C" void kernel_launch(void* const* d_in, const int* in_sizes, int n_in,
                              void* d_out, int out_size, void* d_ws, size_t ws_size,
                              hipStream_t stream)
{
    (void)in_sizes; (void)n_in; (void)out_size; (void)ws_size;

    // --- inputs (setup_inputs dict order, params flattened in insertion order)
    const float* x_in       = (const float*)d_in[0];   // [16384, 64]
    const int*   edge_index = (const int*)  d_in[1];   // [2, 262144]
    const float* edge_attr  = (const float*)d_in[2];   // [262144, 16]
    const int*   node_index = (const int*)  d_in[3];   // [8192]
    const float* mol_fc     = (const float*)d_in[4];   // [512]
    const float* cen_fc     = (const float*)d_in[5];   // [512]
    const int*   batch_idx  = (const int*)  d_in[6];   // [16384]
    // params: convs(3 x {Wl,bl,Wr,We,att,bias}), transfs(3 x {W,b}),
    //         bns(3 x {g,b,mean,var}), gate{W,b}, mlp(3 x {W,b}), out{W,b}
    const int P = 7;

    // --- workspace carve-up (floats, 256B aligned chunks)
    float* ws = (float*)d_ws;
    size_t off = 0;
    auto alloc = [&](size_t n) { float* p = ws + off; off += (n + 63) & ~(size_t)63; return p; };
    float*    xl     = alloc((size_t)N_NODES * HC);
    float*    xr     = alloc((size_t)N_NODES * HC);
    float*    acc    = alloc((size_t)N_NODES * HC);
    float*    xn     = alloc((size_t)N_NODES * EMB);
    float*    logits = alloc((size_t)N_EDGES * HEADS);
    float*    alpha  = alloc((size_t)N_EDGES * HEADS);
    unsigned* nmax   = (unsigned*)alloc((size_t)N_NODES * HEADS);
    float*    nden   = alloc((size_t)N_NODES * HEADS);
    float*    gval   = alloc(N_NODES / 2);
    float*    gex    = alloc(N_NODES / 2);
    unsigned* gmax   = (unsigned*)alloc(N_GRAPHS);
    float*    gden   = alloc(N_GRAPHS);
    float*    hbuf   = alloc((size_t)N_GRAPHS * (EMB + 2));
    float*    m1     = alloc((size_t)N_GRAPHS * DENSE);
    float*    m2     = alloc((size_t)N_GRAPHS * (DENSE / 4));
    float*    m3     = alloc((size_t)N_GRAPHS * (DENSE / 16));

    const float* Xcur = x_in;
    int K = FEAT;
    for (int l = 0; l < 3; ++l) {
        const float* Wl    = (const float*)d_in[P + l * 6 + 0];
        const float* bl    = (const float*)d_in[P + l * 6 + 1];
        const float* Wr    = (const float*)d_in[P + l * 6 + 2];
        const float* We    = (const float*)d_in[P + l * 6 + 3];
        const float* att   = (const float*)d_in[P + l * 6 + 4];
        const float* cbias = (const float*)d_in[P + l * 6 + 5];
        const float* Wt    = (const float*)d_in[P + 18 + l * 2 + 0];
        const float* bt    = (const float*)d_in[P + 18 + l * 2 + 1];
        const float* bng   = (const float*)d_in[P + 24 + l * 4 + 0];
        const float* bnb   = (const float*)d_in[P + 24 + l * 4 + 1];
        const float* bnm   = (const float*)d_in[P + 24 + l * 4 + 2];
        const float* bnv   = (const float*)d_in[P + 24 + l * 4 + 3];

        const size_t shA = 16 * (size_t)(K + 4) * sizeof(float);

        // xl = X @ Wl + bl ; xr = X @ Wr   (WMMA fp32, NT=4 -> 256 cols/block)
        wmma_gemm_f32<4><<<dim3(N_NODES / 16, HC / 256), 128, shA, stream>>>(
            Xcur, Wl, xl, N_NODES, HC, K, bl, nullptr, nullptr, nullptr, nullptr, 1);
        wmma_gemm_f32<4><<<dim3(N_NODES / 16, HC / 256), 128, shA, stream>>>(
            Xcur, Wr, xr, N_NODES, HC, K, nullptr, nullptr, nullptr, nullptr, nullptr, 0);

        init_layer_kernel<<<(N_NODES * HC) / 256, 256, 0, stream>>>(acc, cbias, nmax, nden);
        edge_logits_kernel<<<N_EDGES / 8, 256, 0, stream>>>(
            edge_index, edge_attr, xl, xr, We, att, logits, nmax);
        edge_exp_kernel<<<(N_EDGES * HEADS) / 256, 256, 0, stream>>>(
            edge_index, logits, nmax, alpha, nden);
        edge_scatter_kernel<<<N_EDGES / 8, 256, 0, stream>>>(
            edge_index, xl, alpha, nden, acc);

        // xn = bn(relu(acc @ Wt + bt))   (WMMA fp32, NT=2 -> 128 cols/block)
        const size_t shT = 16 * (size_t)(HC + 4) * sizeof(float);
        wmma_gemm_f32<2><<<dim3(N_NODES / 16, EMB / 128), 128, shT, stream>>>(
            acc, Wt, xn, N_NODES, EMB, HC, bt, bng, bnb, bnm, bnv, 2);

        Xcur = xn;
        K = EMB;
    }

    // --- gated pooling over selected nodes -> hbuf[512, 130]
    const float* gateW = (const float*)d_in[P + 36];
    const float* gateB = (const float*)d_in[P + 37];
    pool_init_kernel<<<(N_GRAPHS * (EMB + 2) + 255) / 256, 256, 0, stream>>>(
        hbuf, mol_fc, cen_fc, gmax, gden);
    gate_kernel<<<(N_NODES / 2) / 8, 256, 0, stream>>>(
        Xcur, node_index, batch_idx, gateW, gateB, gval, gmax);
    gate_exp_kernel<<<(N_NODES / 2) / 256, 256, 0, stream>>>(
        node_index, batch_idx, gval, gmax, gex, gden);
    pool_scatter_kernel<<<(N_NODES / 2) / 8, 256, 0, stream>>>(
        Xcur, node_index, batch_idx, gex, gden, hbuf);

    // --- MLP head (tiny: plain VALU kernels; K=130 breaks WMMA K%4 anyway)
    const float* W0 = (const float*)d_in[P + 38]; const float* b0 = (const float*)d_in[P + 39];
    const float* W1 = (const float*)d_in[P + 40]; const float* b1 = (const float*)d_in[P + 41];
    const float* W2 = (const float*)d_in[P + 42]; const float* b2 = (const float*)d_in[P + 43];
    const float* Wo = (const float*)d_in[P + 44]; const float* bo = (const float*)d_in[P + 45];

    fc_kernel<<<(N_GRAPHS * DENSE + 255) / 256, 256, 0, stream>>>(
        hbuf, W0, b0, m1, N_GRAPHS, DENSE, EMB + 2, 1);
    fc_kernel<<<(N_GRAPHS * (DENSE / 4) + 255) / 256, 256, 0, stream>>>(
        m1, W1, b1, m2, N_GRAPHS, DENSE / 4, DENSE, 1);
    fc_kernel<<<(N_GRAPHS * (DENSE / 16) + 255) / 256, 256, 0, stream>>>(
        m2, W2, b2, m3, N_GRAPHS, DENSE / 16, DENSE / 4, 1);
    fc_kernel<<<(N_GRAPHS + 255) / 256, 256, 0, stream>>>(
        m3, Wo, bo, (float*)d_out, N_GRAPHS, 1, DENSE / 16, 0);
}